// GLQuasiAlphaLowRank_63934883168447
// MI455X (gfx1250) — compile-verified
//
#include <hip/hip_runtime.h>
#include <math.h>

typedef __attribute__((ext_vector_type(2))) float v2f;
typedef __attribute__((ext_vector_type(8))) float v8f;

#define LNUM 4
#define TNUM 8
#define BNUM 32
#define NROW 65536
#define FEAT_IN 224
#define OUTDIM 4096
#define HPAD 257  // 256-bin histogram padded so copies land in distinct banks

// ---------- order-preserving float <-> uint transform ----------
__device__ __forceinline__ unsigned ford(float f) {
  unsigned u = __float_as_uint(f);
  return (u & 0x80000000u) ? ~u : (u | 0x80000000u);
}
__device__ __forceinline__ float iord(unsigned u) {
  return __uint_as_float((u & 0x80000000u) ? (u & 0x7fffffffu) : ~u);
}

// ---------- WMMA f32 16x16x4 helpers ----------
__device__ __forceinline__ v8f wmma4(v2f a, v2f b, v8f c) {
  return __builtin_amdgcn_wmma_f32_16x16x4_f32(false, a, false, b, (short)0, c,
                                               false, false);
}
// A (16x4, row-major, leading dim ld): lanes 0-15 hold rows, VGPR0=K(2*half), VGPR1=K(2*half+1)
__device__ __forceinline__ v2f loadA(const float* A, int ld, int m0, int k0,
                                     int l16, int half) {
  const float* p = A + (m0 + l16) * ld + k0 + 2 * half;
  v2f a;
  a.x = p[0];
  a.y = p[1];
  return a;
}
// B (4x16, row-major, leading dim ld): mirrored layout
__device__ __forceinline__ v2f loadB(const float* Bm, int ld, int k0, int n0,
                                     int l16, int half) {
  const float* p = Bm + (k0 + 2 * half) * ld + n0 + l16;
  v2f b;
  b.x = p[0];
  b.y = p[ld];
  return b;
}

// =====================================================================
// Kernel 1: per-row (l,t,b) stats over N=65536: mean, unbiased var,
// exact quantiles via 4-level 8-bit radix select.  One workgroup/row.
// Level-1 histogram privatized per wave (8 copies, bank-padded) to kill
// same-address LDS atomic serialization on the hot exponent bins.
// Refinement passes re-read the row from L2 (row = 256KB, L2 = 192MB).
// =====================================================================
__global__ __launch_bounds__(256) void stats_kernel(
    const float* __restrict__ wsfeat, float* __restrict__ xfeat) {
  __shared__ unsigned s_wh[8 * HPAD];  // per-wave level-1 hist
  __shared__ unsigned s_rh[6 * HPAD];  // refinement hists (also merged L1)
  __shared__ float s_red[256];
  __shared__ unsigned s_sel[6], s_rem[6];
  __shared__ unsigned s_umin, s_umax;
  __shared__ float s_sum, s_sumsq;

  const int tid = threadIdx.x;
  const int wave = tid >> 5;
  const int r = blockIdx.x;  // r = ((l*T + t)*B + b)
  const float4* row4 =
      reinterpret_cast<const float4*>(wsfeat + (size_t)r * NROW);

  for (int i = tid; i < 8 * HPAD; i += 256) s_wh[i] = 0u;
  if (tid == 0) {
    s_umin = 0xffffffffu;
    s_umax = 0u;
  }
  __syncthreads();

  // ---- Pass 0: stream row (HBM), sum/sumsq/min/max + top-8-bit hist
  float sum = 0.f, sumsq = 0.f;
  unsigned umin = 0xffffffffu, umax = 0u;
  unsigned* wh = &s_wh[wave * HPAD];
  for (int i = tid; i < NROW / 4; i += 256) {
    if (i + 4096 < NROW / 4) __builtin_prefetch(&row4[i + 4096], 0, 0);
    float4 v = row4[i];
    float cs[4] = {v.x, v.y, v.z, v.w};
#pragma unroll
    for (int c = 0; c < 4; ++c) {
      float f = cs[c];
      sum += f;
      sumsq += f * f;
      unsigned u = ford(f);
      umin = (u < umin) ? u : umin;
      umax = (u > umax) ? u : umax;
      atomicAdd(&wh[u >> 24], 1u);
    }
  }
  atomicMin(&s_umin, umin);
  atomicMax(&s_umax, umax);

  // deterministic tree reductions for sum / sumsq
  s_red[tid] = sum;
  __syncthreads();
  for (int s = 128; s > 0; s >>= 1) {
    if (tid < s) s_red[tid] += s_red[tid + s];
    __syncthreads();
  }
  if (tid == 0) s_sum = s_red[0];
  __syncthreads();
  s_red[tid] = sumsq;
  __syncthreads();
  for (int s = 128; s > 0; s >>= 1) {
    if (tid < s) s_red[tid] += s_red[tid + s];
    __syncthreads();
  }
  if (tid == 0) s_sumsq = s_red[0];

  // merge per-wave hists into s_rh[bin] (one thread per bin)
  __syncthreads();
  {
    unsigned tot = 0;
#pragma unroll
    for (int w = 0; w < 8; ++w) tot += s_wh[w * HPAD + tid];
    if (tid < 256) s_rh[tid] = tot;
  }
  __syncthreads();

  // ---- Level-0 scan: top byte + residual rank for 6 interior ranks
  if (tid == 0) {
    const unsigned ranks[6] = {16383u, 16384u, 32767u, 32768u, 49151u, 49152u};
    unsigned cum = 0;
    int j = 0;
    for (int bin = 0; bin < 256 && j < 6; ++bin) {
      unsigned c = s_rh[bin];
      while (j < 6 && ranks[j] < cum + c) {
        s_sel[j] = (unsigned)bin;
        s_rem[j] = ranks[j] - cum;
        ++j;
      }
      cum += c;
    }
  }
  __syncthreads();

  // ---- Levels 1..3: refine 8 bits per pass (rows hit L2)
  for (int shift = 16; shift >= 0; shift -= 8) {
    unsigned selreg[6];
#pragma unroll
    for (int j = 0; j < 6; ++j) selreg[j] = s_sel[j];
    for (int i = tid; i < 6 * HPAD; i += 256) s_rh[i] = 0u;
    __syncthreads();

    for (int i = tid; i < NROW / 4; i += 256) {
      float4 v = row4[i];
      float cs[4] = {v.x, v.y, v.z, v.w};
#pragma unroll
      for (int c = 0; c < 4; ++c) {
        unsigned u = ford(cs[c]);
        unsigned pref = u >> (shift + 8);
        unsigned dig = (u >> shift) & 255u;
#pragma unroll
        for (int j = 0; j < 6; ++j)
          if (pref == selreg[j]) atomicAdd(&s_rh[j * HPAD + dig], 1u);
      }
    }
    __syncthreads();
    if (tid < 6) {
      unsigned rr = s_rem[tid], cum = 0;
      for (int bin = 0; bin < 256; ++bin) {
        unsigned c = s_rh[tid * HPAD + bin];
        if (rr < cum + c) {
          s_sel[tid] = (s_sel[tid] << 8) | (unsigned)bin;
          s_rem[tid] = rr - cum;
          break;
        }
        cum += c;
      }
    }
    __syncthreads();
  }

  if (tid == 0) {
    const float Nf = 65536.f;
    float mean = s_sum / Nf;
    float var = (s_sumsq - s_sum * s_sum / Nf) / (Nf - 1.f);
    float f0 = iord(s_sel[0]), f1 = iord(s_sel[1]);
    float f2 = iord(s_sel[2]), f3 = iord(s_sel[3]);
    float f4 = iord(s_sel[4]), f5 = iord(s_sel[5]);
    float q0 = iord(s_umin), q4 = iord(s_umax);
    float q1 = f0 + 0.75f * (f1 - f0);  // index 0.25*(N-1) = 16383.75
    float q2 = 0.5f * (f2 + f3);        // index 32767.5
    float q3 = f4 + 0.25f * (f5 - f4);  // index 49151.25
    int l = r >> 8, t = (r >> 5) & 7, b = r & 31;
    float* o = xfeat + b * FEAT_IN + (l * TNUM + t) * 7;
    o[0] = mean;
    o[1] = var;
    o[2] = q0;
    o[3] = q1;
    o[4] = q2;
    o[5] = q3;
    o[6] = q4;
  }
}

// =====================================================================
// Kernel 2: feature net: x[32,224] @ W1[224,64] + b1 -> LN -> GELU(erf)
//           -> @ W2[64,32] + b2 -> feat[32,32].  WMMA f32 16x16x4.
// =====================================================================
__global__ __launch_bounds__(256) void featnet_kernel(
    const float* __restrict__ xf, const float* __restrict__ W1,
    const float* __restrict__ b1, const float* __restrict__ g,
    const float* __restrict__ be, const float* __restrict__ W2,
    const float* __restrict__ b2, float* __restrict__ feat) {
  __shared__ float sh1[32 * 72];
  const int tid = threadIdx.x;
  const int wave = tid >> 5, lane = tid & 31;
  const int l16 = lane & 15, half = lane >> 4;

  {  // GEMM1: [32x224]@[224x64], 8 waves = 2x4 tiles of 16x16
    const int mi = wave >> 2, ni = wave & 3;
    const int m0 = mi * 16, n0 = ni * 16;
    v8f acc = {};
    for (int k0 = 0; k0 < FEAT_IN; k0 += 4) {
      v2f a = loadA(xf, FEAT_IN, m0, k0, l16, half);
      v2f b = loadB(W1, 64, k0, n0, l16, half);
      acc = wmma4(a, b, acc);
    }
    float bias = b1[n0 + l16];
#pragma unroll
    for (int v = 0; v < 8; ++v)
      sh1[(m0 + v + 8 * half) * 72 + n0 + l16] = acc[v] + bias;
  }
  __syncthreads();
  if (tid < 32) {  // LayerNorm (biased var) + exact GELU per batch row
    float mu = 0.f;
    for (int j = 0; j < 64; ++j) mu += sh1[tid * 72 + j];
    mu *= (1.f / 64.f);
    float var = 0.f;
    for (int j = 0; j < 64; ++j) {
      float d = sh1[tid * 72 + j] - mu;
      var += d * d;
    }
    var *= (1.f / 64.f);
    float inv = rsqrtf(var + 1e-5f);
    for (int j = 0; j < 64; ++j) {
      float h = (sh1[tid * 72 + j] - mu) * inv * g[j] + be[j];
      h = 0.5f * h * (1.f + erff(h * 0.70710678118654752f));
      sh1[tid * 72 + j] = h;
    }
  }
  __syncthreads();
  if (wave < 4) {  // GEMM2: [32x64]@[64x32], 4 waves = 2x2 tiles
    const int mi = wave >> 1, ni = wave & 1;
    const int m0 = mi * 16, n0 = ni * 16;
    v8f acc = {};
    for (int k0 = 0; k0 < 64; k0 += 4) {
      v2f a = loadA(sh1, 72, m0, k0, l16, half);
      v2f b = loadB(W2, 32, k0, n0, l16, half);
      acc = wmma4(a, b, acc);
    }
    float bias = b2[n0 + l16];
#pragma unroll
    for (int v = 0; v < 8; ++v)
      feat[(m0 + v + 8 * half) * 32 + n0 + l16] = acc[v] + bias;
  }
}

// =====================================================================
// Kernel 3: U/V projections: feat[32,32] @ W_l[32,4096] + b, for
// 4 weight matrices x 4 layers.  One 16x16 WMMA tile per wave.
// =====================================================================
__global__ __launch_bounds__(256) void proj_kernel(
    const float* __restrict__ feat, const float* __restrict__ MUW,
    const float* __restrict__ MUb, const float* __restrict__ MVW,
    const float* __restrict__ MVb, const float* __restrict__ NUW,
    const float* __restrict__ NUb, const float* __restrict__ NVW,
    const float* __restrict__ NVb, float* __restrict__ uv) {
  const int tid = threadIdx.x;
  const int wave = tid >> 5, lane = tid & 31;
  const int l16 = lane & 15, half = lane >> 4;
  int gt = blockIdx.x * 8 + wave;  // 8192 tiles total
  int mat = gt >> 11;              // 0=MU,1=MV,2=NU,3=NV
  int rem = gt & 2047;
  int l = rem >> 9;
  int rem2 = rem & 511;
  int mi = rem2 >> 8;   // 2 row tiles (B=32)
  int ni = rem2 & 255;  // 256 col tiles (4096)
  const float* W = (mat == 0) ? MUW : (mat == 1) ? MVW : (mat == 2) ? NUW : NVW;
  const float* Bv = (mat == 0) ? MUb : (mat == 1) ? MVb : (mat == 2) ? NUb : NVb;
  const float* Wl = W + (size_t)l * 32 * OUTDIM;
  const int m0 = mi * 16, n0 = ni * 16;
  v8f acc = {};
#pragma unroll
  for (int k0 = 0; k0 < 32; k0 += 4) {
    v2f a = loadA(feat, 32, m0, k0, l16, half);
    v2f b = loadB(Wl, OUTDIM, k0, n0, l16, half);
    acc = wmma4(a, b, acc);
  }
  float bias = Bv[l * OUTDIM + n0 + l16];
  float* dst = uv + (size_t)mat * (LNUM * BNUM * OUTDIM);
#pragma unroll
  for (int v = 0; v < 8; ++v)
    dst[(size_t)(l * BNUM + m0 + v + 8 * half) * OUTDIM + n0 + l16] =
        acc[v] + bias;
}

// =====================================================================
// Kernel 4: per (mat,l,b,h): M = U[64,8] @ V[8,64] + 0.1*I.
// 16 tiles of 16x16 per block (8 waves x 2 tiles), K=8 (two wmma steps).
// =====================================================================
__global__ __launch_bounds__(256) void lowrank_kernel(
    const float* __restrict__ uv, float* __restrict__ out) {
  const int tid = threadIdx.x;
  const int wave = tid >> 5, lane = tid & 31;
  const int l16 = lane & 15, half = lane >> 4;
  int bid = blockIdx.x;
  int mat2 = bid >> 10;  // 0 = all_M, 1 = all_N
  int rem = bid & 1023;
  int l = rem >> 8;
  int b = (rem >> 3) & 31;
  int h = rem & 7;
  const size_t slice = (size_t)(l * BNUM + b) * OUTDIM + h * 512;
  const float* U = uv + (size_t)(mat2 * 2 + 0) * (LNUM * BNUM * OUTDIM) + slice;
  const float* V = uv + (size_t)(mat2 * 2 + 1) * (LNUM * BNUM * OUTDIM) + slice;
  float* o = out + (size_t)mat2 * (LNUM * BNUM * 8 * 64 * 64) +
             (size_t)((l * BNUM + b) * 8 + h) * 4096;
#pragma unroll
  for (int t = 2 * wave; t < 2 * wave + 2; ++t) {
    int mi = t >> 2, ni = t & 3;
    int m0 = mi * 16, n0 = ni * 16;
    v8f acc = {};
#pragma unroll
    for (int k0 = 0; k0 < 8; k0 += 4) {
      v2f a = loadA(U, 8, m0, k0, l16, half);
      v2f bb = loadB(V, 64, k0, n0, l16, half);
      acc = wmma4(a, bb, acc);
    }
#pragma unroll
    for (int v = 0; v < 8; ++v) {
      int d = m0 + v + 8 * half;
      int e = n0 + l16;
      o[d * 64 + e] = acc[v] + ((d == e) ? 0.1f : 0.f);
    }
  }
}

extern "C" void kernel_launch(void* const* d_in, const int* in_sizes, int n_in,
                              void* d_out, int out_size, void* d_ws,
                              size_t ws_size, hipStream_t stream) {
  (void)in_sizes;
  (void)n_in;
  (void)out_size;
  (void)ws_size;
  const float* wsfeat = (const float*)d_in[0];
  const float* W1 = (const float*)d_in[1];
  const float* b1 = (const float*)d_in[2];
  const float* g = (const float*)d_in[3];
  const float* be = (const float*)d_in[4];
  const float* W2 = (const float*)d_in[5];
  const float* b2 = (const float*)d_in[6];
  const float* MUW = (const float*)d_in[7];
  const float* MUb = (const float*)d_in[8];
  const float* MVW = (const float*)d_in[9];
  const float* MVb = (const float*)d_in[10];
  const float* NUW = (const float*)d_in[11];
  const float* NUb = (const float*)d_in[12];
  const float* NVW = (const float*)d_in[13];
  const float* NVb = (const float*)d_in[14];

  float* wsf = (float*)d_ws;
  float* xfeat = wsf;          // 32*224 floats
  float* feat = wsf + 8192;    // 32*32 floats
  float* uv = wsf + 16384;     // 4 * 4*32*4096 floats (~8 MB)
  float* out = (float*)d_out;  // [all_M | all_N], 2*4194304 f32

  stats_kernel<<<1024, 256, 0, stream>>>(wsfeat, xfeat);
  featnet_kernel<<<1, 256, 0, stream>>>(xfeat, W1, b1, g, be, W2, b2, feat);
  proj_kernel<<<1024, 256, 0, stream>>>(feat, MUW, MUb, MVW, MVb, NUW, NUb,
                                        NVW, NVb, uv);
  lowrank_kernel<<<2048, 256, 0, stream>>>(uv, out);
}